// DynamicTinyOBBAssigner_86921548137240
// MI455X (gfx1250) — compile-verified
//
#include <hip/hip_runtime.h>

#define EPS_F     1e-9f
#define ALPHA_F   0.5f
#define BETA_F    6.0f
#define RADIUS_F  1.5f
#define TOPK_N    10
#define MAXF_N    5

typedef float v2f __attribute__((ext_vector_type(2)));
typedef float v8f __attribute__((ext_vector_type(8)));

// ---------------------------------------------------------------- zero init
__global__ void k_zero(int* cand, int nc, int* pos, int np) {
    int i = blockIdx.x * blockDim.x + threadIdx.x;
    if (i < nc) cand[i] = 0;
    if (i < np) pos[i] = 0;
}

// ---------------------------------------------------------------- pairwise (WMMA)
// grid: (B, A/16), block 256 (8 waves). Wave w handles gt-tile w, w+8, ...
// am[b][a][g] = align * center_weight * is_in * mask ; cand[b][g] += #in-box anchors
__global__ void __launch_bounds__(256)
k_pairwise(const float* __restrict__ ps, const float* __restrict__ pb,
           const float* __restrict__ ap, const int* __restrict__ gl,
           const float* __restrict__ gb, const int* __restrict__ mg,
           float* __restrict__ am, int* __restrict__ cand,
           int A, int G, int C) {
    const int b  = blockIdx.x;
    const int a0 = blockIdx.y * 16;
    const int tid  = threadIdx.x;
    const int wave = tid >> 5;
    const int lane = tid & 31;

    __shared__ float s_ax[16], s_ay[16], s_px[16], s_py[16];
    __shared__ float s_sc[16][8];           // padded score tile (C <= 8)

    if (tid < 16) {
        size_t row = ((size_t)b * A + (a0 + tid));
        s_px[tid] = pb[row * 5 + 0];
        s_py[tid] = pb[row * 5 + 1];
        s_ax[tid] = ap[(size_t)(a0 + tid) * 2 + 0];
        s_ay[tid] = ap[(size_t)(a0 + tid) * 2 + 1];
        for (int c = 0; c < C; ++c) s_sc[tid][c] = ps[row * C + c];
    }
    __syncthreads();

    const bool hi  = (lane >= 16);
    const int  ma  = lane & 15;
    const int  nwv = blockDim.x >> 5;

    // A operands: rank-4 distance encoding [x, y, |v|^2, 1]
    const float px = s_px[ma], py = s_py[ma];
    v2f Apred; Apred.x = hi ? (px * px + py * py) : px;
               Apred.y = hi ? 1.0f               : py;
    const float ax = s_ax[ma], ay = s_ay[ma];
    v2f Aanc;  Aanc.x  = hi ? (ax * ax + ay * ay) : ax;
               Aanc.y  = hi ? 1.0f                : ay;

    for (int gt = wave; gt * 16 < G; gt += nwv) {
        const int n = lane & 15;
        const int g = gt * 16 + n;
        const float* gp = &gb[((size_t)b * G + g) * 5];
        const float gx = gp[0], gy = gp[1], gw = gp[2], gh = gp[3], gth = gp[4];
        const int   lbl = gl[(size_t)b * G + g];
        const int   msk = mg[(size_t)b * G + g];

        const float scale  = sqrtf(gw * gh);
        const float so     = scale + 1e-6f;
        const float s15    = scale * 1.5f + 1e-6f;
        const float inv_ov = 1.0f / (so * so);
        const float inv_cw = 1.0f / (s15 * s15);
        const float cosT = cosf(-gth), sinT = sinf(-gth);
        const float wR = gw * RADIUS_F, hR = gh * RADIUS_F;

        // B operand: [-2gx, -2gy, 1, |g|^2]
        v2f Bm; Bm.x = hi ? 1.0f              : -2.0f * gx;
                Bm.y = hi ? (gx * gx + gy * gy) : -2.0f * gy;

        v8f cz = {};
        // d2[m,n]  = |p_m - g_n|^2 ; c2[m,n] = |ap_m - g_n|^2 (matrix pipe)
        v8f d2 = __builtin_amdgcn_wmma_f32_16x16x4_f32(false, Apred, false, Bm,
                                                       (short)0, cz, false, false);
        v8f c2 = __builtin_amdgcn_wmma_f32_16x16x4_f32(false, Aanc,  false, Bm,
                                                       (short)0, cz, false, false);
        const int mbase = hi ? 8 : 0;
        int cnt = 0;
#pragma unroll
        for (int r = 0; r < 8; ++r) {
            const int m = mbase + r;
            const float axm = s_ax[m], aym = s_ay[m];
            const float scv = s_sc[m][lbl];

            float dd = fmaxf(d2[r], 0.0f);
            float cc = fmaxf(c2[r], 0.0f);
            float ov = expf(-dd * inv_ov);
            float cw = expf(-cc * inv_cw);
            float bs = fmaxf(1.0f / (1.0f + expf(-scv)), EPS_F);
            float alg = expf(ALPHA_F * logf(bs + EPS_F) + BETA_F * logf(ov + EPS_F)) * cw;

            float dx = axm - gx, dy = aym - gy;
            float ldx = dx * cosT - dy * sinT;
            float ldy = dx * sinT + dy * cosT;
            bool isin = (fabsf(ldx) < wR) && (fabsf(ldy) < hR);
            cnt += isin ? 1 : 0;

            float amv = (isin && msk > 0) ? alg : 0.0f;
            am[((size_t)b * A + (a0 + m)) * G + g] = amv;
        }
        if (cnt) atomicAdd(&cand[b * G + g], cnt);
    }
}

// ---------------------------------------------------------------- tiny-GT fallback
// grid: B*G blocks of 256; dynamic LDS = A*4 bytes
__global__ void __launch_bounds__(256)
k_fallback(const float* __restrict__ ps, const float* __restrict__ pb,
           const float* __restrict__ ap, const int* __restrict__ gl,
           const float* __restrict__ gb, const int* __restrict__ mg,
           const int* __restrict__ cand, float* __restrict__ am,
           int A, int G, int C) {
    const int bg = blockIdx.x;
    const int b  = bg / G;
    const float* gp = &gb[(size_t)bg * 5];
    const float gw = gp[2], gh = gp[3];
    const float area = gw * gh;
    const bool tiny = (mg[bg] > 0) && (area < 0.01f) && (cand[bg] == 0);
    if (!tiny) return;

    extern __shared__ float s_dist[];                 // A floats
    __shared__ float s_rv[256];
    __shared__ int   s_ri[256];
    __shared__ int   s_sel[MAXF_N];
    const int tid = threadIdx.x;
    const float gx = gp[0], gy = gp[1];

    for (int a = tid; a < A; a += 256) {
        float dx = gx - ap[(size_t)a * 2 + 0];
        float dy = gy - ap[(size_t)a * 2 + 1];
        s_dist[a] = dx * dx + dy * dy;
    }
    __syncthreads();

    for (int r = 0; r < MAXF_N; ++r) {                // 5 closest anchors, stable ties
        float v = 3.4e38f; int vi = 0x7fffffff;
        for (int a = tid; a < A; a += 256) {
            float x = s_dist[a];
            if (x < v) { v = x; vi = a; }
        }
        s_rv[tid] = v; s_ri[tid] = vi;
        __syncthreads();
        for (int s = 128; s > 0; s >>= 1) {
            if (tid < s) {
                float o = s_rv[tid + s]; int oi = s_ri[tid + s];
                if (o < s_rv[tid] || (o == s_rv[tid] && oi < s_ri[tid])) {
                    s_rv[tid] = o; s_ri[tid] = oi;
                }
            }
            __syncthreads();
        }
        if (tid == 0) { s_sel[r] = s_ri[0]; s_dist[s_ri[0]] = 3.4e38f; }
        __syncthreads();
    }

    const int kf = (area < 0.002f) ? 5 : 3;
    if (tid < kf) {
        const int a = s_sel[tid];
        const int lbl = gl[bg];
        size_t row = (size_t)b * A + a;
        float scv = ps[row * C + lbl];
        float bs  = fmaxf(1.0f / (1.0f + expf(-scv)), EPS_F);
        float px = pb[row * 5 + 0], py = pb[row * 5 + 1];
        float scale = sqrtf(gw * gh);
        float so = scale + 1e-6f, s15 = scale * 1.5f + 1e-6f;
        float dx = px - gx, dy = py - gy;
        float ov = expf(-(dx * dx + dy * dy) / (so * so));
        float axv = ap[(size_t)a * 2 + 0], ayv = ap[(size_t)a * 2 + 1];
        float cdx = axv - gx, cdy = ayv - gy;
        float cw = expf(-(cdx * cdx + cdy * cdy) / (s15 * s15));
        float alg = expf(ALPHA_F * logf(bs + EPS_F) + BETA_F * logf(ov + EPS_F)) * cw;
        am[row * G + (bg % G)] = alg;                 // is_in forced true, mask known 1
    }
}

// ---------------------------------------------------------------- top-10 per (b,g)
// grid: B*G blocks of 256; dynamic LDS = A*4 bytes
__global__ void __launch_bounds__(256)
k_topk(const float* __restrict__ pb, const float* __restrict__ gb,
       const float* __restrict__ am, float* __restrict__ am_max,
       int* __restrict__ pos, int A, int G) {
    const int bg = blockIdx.x;
    const int b  = bg / G;
    const int g  = bg % G;
    extern __shared__ float s_row[];                  // A floats
    __shared__ float s_rv[256];
    __shared__ int   s_ri[256];
    __shared__ float s_tv[TOPK_N];
    __shared__ int   s_ti[TOPK_N];
    __shared__ float s_ov[TOPK_N];
    const int tid = threadIdx.x;

    for (int a = tid; a < A; a += 256)
        s_row[a] = am[((size_t)b * A + a) * G + g];
    __syncthreads();

    for (int r = 0; r < TOPK_N; ++r) {                // stable argmax rounds
        float v = -2.0f; int vi = 0x7fffffff;
        for (int a = tid; a < A; a += 256) {
            float x = s_row[a];
            if (x > v) { v = x; vi = a; }
        }
        s_rv[tid] = v; s_ri[tid] = vi;
        __syncthreads();
        for (int s = 128; s > 0; s >>= 1) {
            if (tid < s) {
                float o = s_rv[tid + s]; int oi = s_ri[tid + s];
                if (o > s_rv[tid] || (o == s_rv[tid] && oi < s_ri[tid])) {
                    s_rv[tid] = o; s_ri[tid] = oi;
                }
            }
            __syncthreads();
        }
        if (tid == 0) { s_tv[r] = s_rv[0]; s_ti[r] = s_ri[0]; s_row[s_ri[0]] = -1.0f; }
        __syncthreads();
    }

    if (tid < TOPK_N) {                               // overlaps at winners
        const int a = s_ti[tid];
        size_t row = (size_t)b * A + a;
        float px = pb[row * 5 + 0], py = pb[row * 5 + 1];
        const float* gp = &gb[(size_t)bg * 5];
        float scale = sqrtf(gp[2] * gp[3]);
        float so = scale + 1e-6f;
        float dx = px - gp[0], dy = py - gp[1];
        s_ov[tid] = expf(-(dx * dx + dy * dy) / (so * so));
    }
    __syncthreads();
    if (tid == 0) {
        float s = 0.0f;
        for (int i = 0; i < TOPK_N; ++i) s += s_ov[i];
        float k = fminf(fmaxf(rintf(s), 1.0f), (float)TOPK_N);   // banker's rounding
        am_max[bg] = s_tv[0];
        for (int i = 0; i < TOPK_N; ++i)
            if ((float)i < k && s_tv[i] > EPS_F)
                pos[(size_t)b * A + s_ti[i]] = 1;
    }
}

// ---------------------------------------------------------------- per-anchor outputs
__global__ void __launch_bounds__(256)
k_out(const float* __restrict__ pb, const int* __restrict__ gl,
      const float* __restrict__ gb, const float* __restrict__ am,
      const float* __restrict__ am_max, const int* __restrict__ pos,
      float* __restrict__ out, int B, int A, int G, int C) {
    const int i = blockIdx.x * blockDim.x + threadIdx.x;
    if (i >= B * A) return;
    const int b = i / A;

    const float* r = &am[(size_t)i * G];
    float best = r[0]; int bi = 0;
    for (int g = 1; g < G; ++g) { float x = r[g]; if (x > best) { best = x; bi = g; } }

    const int bg = b * G + bi;
    const int lbl = gl[bg];
    const float* gp = &gb[(size_t)bg * 5];

    float scale = sqrtf(gp[2] * gp[3]);
    float so = scale + 1e-6f;
    float px = pb[(size_t)i * 5 + 0], py = pb[(size_t)i * 5 + 1];
    float dx = px - gp[0], dy = py - gp[1];
    float ov = expf(-(dx * dx + dy * dy) / (so * so));
    float raw = best / (am_max[bg] + EPS_F) * ov;

    const int p = pos[i];
    const bool valid = (lbl >= 0) && (lbl < C) && (p != 0);
    float soft = (raw > 0.0f) ? sqrtf(raw) : 0.0f;    // raw^(1/TEMPERATURE), T=2
    float sv = valid ? soft : 0.0f;

    const size_t BA = (size_t)B * A;
    out[i] = (float)lbl;                              // target_labels
    float* ob = out + BA + (size_t)i * 5;             // target_bboxes
    for (int c = 0; c < 5; ++c) ob[c] = gp[c];
    float* os = out + BA * 6 + (size_t)i * C;         // target_scores (one-hot * sv)
    for (int c = 0; c < C; ++c) os[c] = (c == lbl) ? sv : 0.0f;
    out[BA * 6 + BA * C + i] = (p != 0) ? 1.0f : 0.0f;  // is_pos_mask
}

// ---------------------------------------------------------------- launcher
extern "C" void kernel_launch(void* const* d_in, const int* in_sizes, int n_in,
                              void* d_out, int out_size, void* d_ws, size_t ws_size,
                              hipStream_t stream) {
    const float* ps = (const float*)d_in[0];   // pred_scores  (B,A,C)
    const float* pb = (const float*)d_in[1];   // pred_bboxes  (B,A,5)
    const float* ap = (const float*)d_in[2];   // anchor_points(A,2)
    const int*   gl = (const int*)  d_in[3];   // gt_labels    (B,G,1)
    const float* gb = (const float*)d_in[4];   // gt_bboxes    (B,G,5)
    const int*   mg = (const int*)  d_in[5];   // mask_gt      (B,G,1)

    const int C = 5;
    const int A = in_sizes[2] / 2;
    const int B = in_sizes[0] / (A * C);
    const int G = in_sizes[4] / (B * 5);

    float* am     = (float*)d_ws;                       // B*A*G
    float* am_max = am + (size_t)B * A * G;             // B*G
    int*   cand   = (int*)(am_max + (size_t)B * G);     // B*G
    int*   pos    = cand + (size_t)B * G;               // B*A

    const int nmax = (B * A > B * G) ? B * A : B * G;
    k_zero<<<(nmax + 255) / 256, 256, 0, stream>>>(cand, B * G, pos, B * A);

    dim3 g1(B, A / 16);
    k_pairwise<<<g1, 256, 0, stream>>>(ps, pb, ap, gl, gb, mg, am, cand, A, G, C);

    const size_t ldsA = (size_t)A * sizeof(float);
    k_fallback<<<B * G, 256, ldsA, stream>>>(ps, pb, ap, gl, gb, mg, cand, am, A, G, C);
    k_topk    <<<B * G, 256, ldsA, stream>>>(pb, gb, am, am_max, pos, A, G);

    k_out<<<(B * A + 255) / 256, 256, 0, stream>>>(pb, gl, gb, am, am_max, pos,
                                                   (float*)d_out, B, A, G, C);
}